// BraskModel_15745350107407
// MI455X (gfx1250) — compile-verified
//
#include <hip/hip_runtime.h>
#include <hip/hip_bf16.h>

typedef __attribute__((ext_vector_type(2))) float v2f;
typedef __attribute__((ext_vector_type(8))) float v8f;

// Problem constants (match reference)
#define BC 2
#define LC 128
#define HC 768
#define RC 24
#define SC 8
#define AC 256
#define RDC 100

// ---------------------------------------------------------------------------
// Generic tiny fp32 WMMA GEMM: C[m,n] = sum_k A[m,k] * B(k,n) + bias[n]
//   BT=true : B is row-major (N,K) weight matrix -> B(k,n) = Bm[n*K+k]
//   BT=false: B is row-major (K,N) activation    -> B(k,n) = Bm[k*N+n]
// One wave (32 threads) per 16x16 output tile; blockIdx.z = batch with
// element strides sA/sB/sC. Uses V_WMMA_F32_16X16X4_F32 (full fp32, no
// conversion). Requires K % 4 == 0 (true for all call sites: 768/128/100).
// Fragment layout per CDNA5 ISA 7.12.2 (32-bit A 16x4):
//   lane = half*16 + idx ; a[j] = A[row, k0 + 2*half + j], j in {0,1}
//   B mirrored per column: b[j] = B(k0 + 2*half + j, col)
// C/D f32 16x16: element (m,n) -> VGPR m%8, lane n + 16*(m/8).
// ---------------------------------------------------------------------------
template <bool BT>
__global__ void gemm16_wmma(const float* __restrict__ A,
                            const float* __restrict__ Bm,
                            const float* __restrict__ bias,
                            float* __restrict__ C,
                            int M, int N, int K,
                            long sA, long sB, long sC) {
  const int lane = threadIdx.x & 31;
  const int half = lane >> 4;     // 0 | 1
  const int idx  = lane & 15;
  const int tm = blockIdx.y, tn = blockIdx.x, bz = blockIdx.z;

  const float* Ab = A  + (long)bz * sA;
  const float* Bb = Bm + (long)bz * sB;
  float*       Cb = C  + (long)bz * sC;

  int row = tm * 16 + idx; int rc = row < M ? row : (M - 1);
  int col = tn * 16 + idx; int cc = col < N ? col : (N - 1);
  const float* Ar = Ab + (long)rc * K + 2 * half;   // contiguous K pair base
  const float* Br = BT ? (Bb + (long)cc * K + 2 * half) : nullptr;

  v8f acc = {};
#pragma unroll 8
  for (int k0 = 0; k0 < K; k0 += 4) {
    v2f a, b;
    const float2 av = *(const float2*)(Ar + k0);    // A[m, k0+2h .. +1]
    a[0] = av.x; a[1] = av.y;
    if (BT) {
      const float2 bv = *(const float2*)(Br + k0);  // W[n, k0+2h .. +1]
      b[0] = bv.x; b[1] = bv.y;
    } else {
      b[0] = Bb[(long)(k0 + 2 * half) * N + cc];    // B[k, n] strided
      b[1] = Bb[(long)(k0 + 2 * half + 1) * N + cc];
    }
    acc = __builtin_amdgcn_wmma_f32_16x16x4_f32(
        false, a, false, b, (short)0, acc, false, false);
  }

#pragma unroll
  for (int v = 0; v < 8; ++v) {
    int m = tm * 16 + half * 8 + v;
    int n = tn * 16 + idx;
    if (m < M && n < N)
      Cb[(long)m * N + n] = acc[v] + (bias ? bias[n] : 0.f);
  }
}

// ---------------------------------------------------------------------------
// Row-wise double dot: out1[m]=sc*dot(in[m]+add[m], w1)+b1 ; same for out2.
// One wave per row, wave-shuffle reduction.
// ---------------------------------------------------------------------------
__global__ void dot2_kernel(const float* __restrict__ in,
                            const float* __restrict__ add,
                            const float* __restrict__ rowscale,
                            const float* __restrict__ w1,
                            const float* __restrict__ w2,
                            const float* __restrict__ b1,
                            const float* __restrict__ b2,
                            float* __restrict__ out1,
                            float* __restrict__ out2,
                            int M, int K) {
  const int lane = threadIdx.x;
  const int m = blockIdx.x * blockDim.y + threadIdx.y;
  if (m >= M) return;
  const float* p = in + (long)m * K;
  const float* q = add ? add + (long)m * K : nullptr;
  float a1 = 0.f, a2 = 0.f;
  for (int k = lane; k < K; k += 32) {
    float x = p[k] + (q ? q[k] : 0.f);
    a1 += x * w1[k];
    a2 += x * w2[k];
  }
  for (int off = 16; off; off >>= 1) {
    a1 += __shfl_xor(a1, off, 32);
    a2 += __shfl_xor(a2, off, 32);
  }
  if (lane == 0) {
    float sc = rowscale ? rowscale[m] : 1.f;
    out1[m] = sc * a1 + (b1 ? b1[0] : 0.f);
    out2[m] = sc * a2 + (b2 ? b2[0] : 0.f);
  }
}

// ---------------------------------------------------------------------------
// e[b,r,l] = sum_a tanh(wx[b,l,a]+wr[r,a]+wg[b,a])*vw[a] + vb ; masked.
// One wave per (b,r,l).
// ---------------------------------------------------------------------------
__global__ void attn_logits_kernel(const float* __restrict__ wx,
                                   const float* __restrict__ wr,
                                   const float* __restrict__ wg,
                                   const float* __restrict__ vw,
                                   const float* __restrict__ vb,
                                   const float* __restrict__ mask,
                                   float* __restrict__ e) {
  const int lane = threadIdx.x;
  const int p = blockIdx.x * blockDim.y + threadIdx.y;
  if (p >= BC * RC * LC) return;
  const int l = p % LC;
  const int r = (p / LC) % RC;
  const int b = p / (LC * RC);
  const float* px = wx + (long)(b * LC + l) * AC;
  const float* pr = wr + (long)r * AC;
  const float* pg = wg + (long)b * AC;
  float acc = 0.f;
  for (int a = lane; a < AC; a += 32)
    acc += tanhf(px[a] + pr[a] + pg[a]) * vw[a];
  for (int off = 16; off; off >>= 1) acc += __shfl_xor(acc, off, 32);
  if (lane == 0) {
    float ev = acc + vb[0];
    e[p] = (mask[b * LC + l] > 0.f) ? ev : -1e30f;
  }
}

// Softmax over L=128 per (b,r) row; one 128-thread block per row. In-place.
__global__ void softmax_kernel(float* __restrict__ e) {
  __shared__ float red[LC];
  float* p = e + (long)blockIdx.x * LC;
  const int t = threadIdx.x;
  float v = p[t];
  red[t] = v; __syncthreads();
  for (int s = LC / 2; s > 0; s >>= 1) {
    if (t < s) red[t] = fmaxf(red[t], red[t + s]);
    __syncthreads();
  }
  float mx = red[0]; __syncthreads();
  float ex = __expf(v - mx);
  red[t] = ex; __syncthreads();
  for (int s = LC / 2; s > 0; s >>= 1) {
    if (t < s) red[t] += red[t + s];
    __syncthreads();
  }
  p[t] = ex / red[0];
}

// sk[b,s,:] = 0.5*(X[b,starts]+X[b,ends]); also writes sk_mask = 1.
__global__ void gather_sk_kernel(const float* __restrict__ X,
                                 const int* __restrict__ st,
                                 const int* __restrict__ en,
                                 float* __restrict__ sk,
                                 float* __restrict__ skmask) {
  const int i = blockIdx.x * blockDim.x + threadIdx.x;
  if (i < BC * SC) skmask[i] = 1.f;
  if (i >= BC * SC * HC) return;
  const int h = i % HC;
  const int s = (i / HC) % SC;
  const int b = i / (HC * SC);
  int i1 = st[b * SC + s], i2 = en[b * SC + s];
  sk[i] = 0.5f * (X[((long)b * LC + i1) * HC + h] +
                  X[((long)b * LC + i2) * HC + h]);
}

// Broadcast-add the separable terms into the 4 (B,R,S,L) logit outputs.
__global__ void assemble_kernel(const float* PF1, const float* PF2,
                                const float* QF1, const float* QF2,
                                const float* RF1, const float* RF2,
                                const float* PB1, const float* PB2,
                                const float* QB1, const float* QB2,
                                const float* RB1, const float* RB2,
                                float* fts, float* fte,
                                float* bhs, float* bhe) {
  const int i = blockIdx.x * blockDim.x + threadIdx.x;
  if (i >= BC * RC * SC * LC) return;
  const int l = i % LC;
  const int s = (i / LC) % SC;
  const int r = (i / (LC * SC)) % RC;
  const int b = i / (LC * SC * RC);
  const int bs = b * SC + s, bl = b * LC + l, br = b * RC + r;
  fts[i] = PF1[bs] + QF1[bl] + RF1[br];
  fte[i] = PF2[bs] + QF2[bl] + RF2[br];
  bhs[i] = PB1[bs] + QB1[bl] + RB1[br];
  bhe[i] = PB2[bs] + QB2[bl] + RB2[br];
}

// ---------------------------------------------------------------------------
extern "C" void kernel_launch(void* const* d_in, const int* in_sizes, int n_in,
                              void* d_out, int out_size, void* d_ws, size_t ws_size,
                              hipStream_t stream) {
  (void)in_sizes; (void)n_in; (void)out_size; (void)ws_size;
  // ---- inputs (setup_inputs dict order) ----
  const float* X      = (const float*)d_in[0];
  const float* Xmean  = (const float*)d_in[1];
  const float* mask   = (const float*)d_in[2];
  const float* semrel = (const float*)d_in[3];
  const float* trerel = (const float*)d_in[4];
  const int*   sstart = (const int*)d_in[5];
  const int*   send   = (const int*)d_in[6];
  const float *fh_ws=(const float*)d_in[7],  *fh_bs=(const float*)d_in[8],
              *fh_we=(const float*)d_in[9],  *fh_be=(const float*)d_in[10];
  const float *bt_ws=(const float*)d_in[11], *bt_bs=(const float*)d_in[12],
              *bt_we=(const float*)d_in[13], *bt_be=(const float*)d_in[14];
  const float *ft_ws=(const float*)d_in[15], *ft_bs=(const float*)d_in[16],
              *ft_we=(const float*)d_in[17], *ft_be=(const float*)d_in[18];
  const float *bh_ws=(const float*)d_in[19], *bh_bs=(const float*)d_in[20],
              *bh_we=(const float*)d_in[21], *bh_be=(const float*)d_in[22];
  const float *fra_wr_w=(const float*)d_in[23], *fra_wr_b=(const float*)d_in[24],
              *fra_wg_w=(const float*)d_in[25], *fra_wg_b=(const float*)d_in[26],
              *fra_wx_w=(const float*)d_in[27], *fra_wx_b=(const float*)d_in[28],
              *fra_v_w =(const float*)d_in[29], *fra_v_b =(const float*)d_in[30];
  const float *bra_wr_w=(const float*)d_in[31], *bra_wr_b=(const float*)d_in[32],
              *bra_wg_w=(const float*)d_in[33], *bra_wg_b=(const float*)d_in[34],
              *bra_wx_w=(const float*)d_in[35], *bra_wx_b=(const float*)d_in[36],
              *bra_v_w =(const float*)d_in[37], *bra_v_b =(const float*)d_in[38];
  const float *ffe_ws_w=(const float*)d_in[39], *ffe_ws_b=(const float*)d_in[40],
              *ffe_wx_w=(const float*)d_in[41], *ffe_wx_b=(const float*)d_in[42];
  const float *bfe_ws_w=(const float*)d_in[43], *bfe_ws_b=(const float*)d_in[44],
              *bfe_wx_w=(const float*)d_in[45], *bfe_wx_b=(const float*)d_in[46];

  // ---- outputs (concat, return order) ----
  float* out = (float*)d_out;
  float* o_fhs  = out + 0;
  float* o_fhe  = out + 256;
  float* o_bts  = out + 512;
  float* o_bte  = out + 768;
  float* o_fts  = out + 1024;
  float* o_fte  = out + 1024 + 49152;
  float* o_bhs2 = out + 1024 + 2 * 49152;
  float* o_bhe2 = out + 1024 + 3 * 49152;
  float* o_sk   = out + 1024 + 4 * 49152;            // (B,S,H)
  float* o_skm  = o_sk + BC * SC * HC;               // (B,S)

  // ---- workspace carve (floats) ----
  float* w = (float*)d_ws;
  long o = 0;
  float* wxF   = w + o; o += (long)BC*LC*AC;   // 65536
  float* wxB   = w + o; o += (long)BC*LC*AC;
  float* wrF   = w + o; o += (long)RC*AC;      // 6144
  float* wrB   = w + o; o += (long)RC*AC;
  float* wgF   = w + o; o += (long)BC*AC;      // 512
  float* wgB   = w + o; o += (long)BC*AC;
  float* eF    = w + o; o += (long)BC*RC*LC;   // 6144 (becomes attn)
  float* eB    = w + o; o += (long)BC*RC*LC;
  float* cF    = w + o; o += (long)BC*RC*HC;   // 36864
  float* cB    = w + o; o += (long)BC*RC*HC;
  float* wskF  = w + o; o += (long)BC*SC*HC;   // 12288
  float* wskB  = w + o; o += (long)BC*SC*HC;
  float* wxFFE = w + o; o += (long)BC*LC*HC;   // 196608
  float* wxBFE = w + o; o += (long)BC*LC*HC;
  float* PF1 = w + o; o += BC*SC;  float* PF2 = w + o; o += BC*SC;
  float* QF1 = w + o; o += BC*LC;  float* QF2 = w + o; o += BC*LC;
  float* RF1 = w + o; o += BC*RC;  float* RF2 = w + o; o += BC*RC;
  float* PB1 = w + o; o += BC*SC;  float* PB2 = w + o; o += BC*SC;
  float* QB1 = w + o; o += BC*LC;  float* QB2 = w + o; o += BC*LC;
  float* RB1 = w + o; o += BC*RC;  float* RB2 = w + o; o += BC*RC;

  const dim3 wv(32, 1, 1);       // one wave
  const dim3 d2(32, 8, 1);       // 8 waves (dot2/attn-logits)

  // 1) sk gather + sk_mask (outputs)
  gather_sk_kernel<<<(BC*SC*HC + 255) / 256, 256, 0, stream>>>(
      X, sstart, send, o_sk, o_skm);

  // 2) first-stage entity logits (B,L)
  dot2_kernel<<<BC*LC/8, d2, 0, stream>>>(X, nullptr, nullptr, fh_ws, fh_we,
                                          fh_bs, fh_be, o_fhs, o_fhe, BC*LC, HC);
  dot2_kernel<<<BC*LC/8, d2, 0, stream>>>(X, nullptr, nullptr, bt_ws, bt_we,
                                          bt_bs, bt_be, o_bts, o_bte, BC*LC, HC);

  // 3) attention projections (WMMA GEMMs, weights are (A,K) row-major -> BT)
  gemm16_wmma<true><<<dim3(AC/16, BC*LC/16, 1), wv, 0, stream>>>(
      X, fra_wx_w, fra_wx_b, wxF, BC*LC, AC, HC, 0, 0, 0);
  gemm16_wmma<true><<<dim3(AC/16, (RC+15)/16, 1), wv, 0, stream>>>(
      semrel, fra_wr_w, fra_wr_b, wrF, RC, AC, HC, 0, 0, 0);
  gemm16_wmma<true><<<dim3(AC/16, 1, 1), wv, 0, stream>>>(
      Xmean, fra_wg_w, fra_wg_b, wgF, BC, AC, HC, 0, 0, 0);
  gemm16_wmma<true><<<dim3(AC/16, BC*LC/16, 1), wv, 0, stream>>>(
      X, bra_wx_w, bra_wx_b, wxB, BC*LC, AC, HC, 0, 0, 0);
  gemm16_wmma<true><<<dim3(AC/16, (RC+15)/16, 1), wv, 0, stream>>>(
      trerel, bra_wr_w, bra_wr_b, wrB, RC, AC, RDC, 0, 0, 0);  // K=100 (4|100)
  gemm16_wmma<true><<<dim3(AC/16, 1, 1), wv, 0, stream>>>(
      Xmean, bra_wg_w, bra_wg_b, wgB, BC, AC, HC, 0, 0, 0);

  // 4) tanh-MLP attention scores (masked)
  attn_logits_kernel<<<(BC*RC*LC + 7) / 8, d2, 0, stream>>>(
      wxF, wrF, wgF, fra_v_w, fra_v_b, mask, eF);
  attn_logits_kernel<<<(BC*RC*LC + 7) / 8, d2, 0, stream>>>(
      wxB, wrB, wgB, bra_v_w, bra_v_b, mask, eB);

  // 5) softmax over L, in place
  softmax_kernel<<<BC*RC, LC, 0, stream>>>(eF);
  softmax_kernel<<<BC*RC, LC, 0, stream>>>(eB);

  // 6) context c[b,r,:] = attn[b] @ X[b]  (batched, B operand is (K,N))
  gemm16_wmma<false><<<dim3(HC/16, (RC+15)/16, BC), wv, 0, stream>>>(
      eF, X, nullptr, cF, RC, HC, LC, (long)RC*LC, (long)LC*HC, (long)RC*HC);
  gemm16_wmma<false><<<dim3(HC/16, (RC+15)/16, BC), wv, 0, stream>>>(
      eB, X, nullptr, cB, RC, HC, LC, (long)RC*LC, (long)LC*HC, (long)RC*HC);

  // 7) fuse projections (WMMA GEMMs)
  gemm16_wmma<true><<<dim3(HC/16, BC*SC/16, 1), wv, 0, stream>>>(
      o_sk, ffe_ws_w, ffe_ws_b, wskF, BC*SC, HC, HC, 0, 0, 0);
  gemm16_wmma<true><<<dim3(HC/16, BC*SC/16, 1), wv, 0, stream>>>(
      o_sk, bfe_ws_w, bfe_ws_b, wskB, BC*SC, HC, HC, 0, 0, 0);
  gemm16_wmma<true><<<dim3(HC/16, BC*LC/16, 1), wv, 0, stream>>>(
      X, ffe_wx_w, ffe_wx_b, wxFFE, BC*LC, HC, HC, 0, 0, 0);
  gemm16_wmma<true><<<dim3(HC/16, BC*LC/16, 1), wv, 0, stream>>>(
      X, bfe_wx_w, bfe_wx_b, wxBFE, BC*LC, HC, HC, 0, 0, 0);

  // 8) separable head dots: P (per b,s; carries the head bias + sk_mask),
  //    Q (per b,l; includes +X), R (per b,r)
  dot2_kernel<<<BC*SC/8, d2, 0, stream>>>(wskF, nullptr, o_skm, ft_ws, ft_we,
                                          ft_bs, ft_be, PF1, PF2, BC*SC, HC);
  dot2_kernel<<<BC*LC/8, d2, 0, stream>>>(wxFFE, X, nullptr, ft_ws, ft_we,
                                          nullptr, nullptr, QF1, QF2, BC*LC, HC);
  dot2_kernel<<<BC*RC/8, d2, 0, stream>>>(cF, nullptr, nullptr, ft_ws, ft_we,
                                          nullptr, nullptr, RF1, RF2, BC*RC, HC);
  dot2_kernel<<<BC*SC/8, d2, 0, stream>>>(wskB, nullptr, o_skm, bh_ws, bh_we,
                                          bh_bs, bh_be, PB1, PB2, BC*SC, HC);
  dot2_kernel<<<BC*LC/8, d2, 0, stream>>>(wxBFE, X, nullptr, bh_ws, bh_we,
                                          nullptr, nullptr, QB1, QB2, BC*LC, HC);
  dot2_kernel<<<BC*RC/8, d2, 0, stream>>>(cB, nullptr, nullptr, bh_ws, bh_we,
                                          nullptr, nullptr, RB1, RB2, BC*RC, HC);

  // 9) broadcast assemble the 4 (B,R,S,L) outputs
  assemble_kernel<<<(BC*RC*SC*LC + 255) / 256, 256, 0, stream>>>(
      PF1, PF2, QF1, QF2, RF1, RF2, PB1, PB2, QB1, QB2, RB1, RB2,
      o_fts, o_fte, o_bhs2, o_bhe2);
}